// FactorizedVectorQuantize_34926674051496
// MI455X (gfx1250) — compile-verified
//
#include <hip/hip_runtime.h>
#include <hip/hip_bf16.h>
#include <math.h>
#include <float.h>
#include <limits.h>

typedef __attribute__((ext_vector_type(2))) float v2f;
typedef __attribute__((ext_vector_type(8))) float v8f;
typedef __attribute__((ext_vector_type(4))) unsigned v4u;
typedef __attribute__((ext_vector_type(8))) unsigned v8u;

// Problem sizes
#define BB   16
#define DIN  1024
#define TT   1024
#define KK   8192
#define DCB  256
#define NTOK (BB * TT)          // 16384 rows of enc

// d_out layout (floats, reference tuple order)
#define OFF_ZQOUT  0ull
#define OFF_IDX    16777216ull
#define OFF_COMMIT 16793600ull
#define OFF_CBLOSS 16793616ull
#define OFF_DIST   16793632ull
#define OFF_PERP   151011360ull
#define OFF_ACT    151011361ull

// workspace layout (float offsets)
#define WS_WIN   0ull            // 256*1024
#define WS_WOUT  262144ull       // 1024*256
#define WS_CBN   524288ull       // 8192*256
#define WS_CBSQ  2621440ull      // 8192
#define WS_ENC   2629632ull      // z_e in (bt, c) layout, 16384*256
#define WS_ENCN  6823936ull      // normalized enc, 16384*256
#define WS_ENCSQ 11018240ull     // 16384
#define WS_ZQ    11034624ull     // z_q in (bt, c) layout, 16384*256
#define WS_MSE   15228928ull     // 16 floats
#define WS_IDX   15228944ull     // 16384 ints
#define WS_CNT   15245328ull     // 8192 ints

__device__ __forceinline__ float warp_sum(float v) {
  for (int o = 16; o; o >>= 1) v += __shfl_xor(v, o, 32);
  return v;
}

// blockDim.x == 256 assumed (8 waves of 32)
__device__ __forceinline__ float block_sum(float v, float* s) {
  __syncthreads();
  int lane = threadIdx.x & 31, w = threadIdx.x >> 5;
  v = warp_sum(v);
  if (lane == 0) s[w] = v;
  __syncthreads();
  float r = (threadIdx.x < 8) ? s[threadIdx.x] : 0.0f;
  if (w == 0) {
    r = warp_sum(r);
    if (lane == 0) s[0] = r;
  }
  __syncthreads();
  return s[0];
}

// ---------------------------------------------------------------- zero
__global__ void vq_zero(int* cnt, float* mse) {
  int i = blockIdx.x * 256 + threadIdx.x;
  if (i < KK) cnt[i] = 0;
  if (i < BB) mse[i] = 0.0f;
}

// ------------------------------------------------- weight-norm weights
__global__ void vq_prep_weights(const float* in_g, const float* in_v,
                                const float* out_g, const float* out_v,
                                float* w_in, float* w_out) {
  __shared__ float s[8];
  int row = blockIdx.x;
  const float* v;
  const float* g;
  float* w;
  int len;
  if (row < DCB) { v = in_v + (size_t)row * DIN; g = in_g; w = w_in + (size_t)row * DIN; len = DIN; }
  else { row -= DCB; v = out_v + (size_t)row * DCB; g = out_g; w = w_out + (size_t)row * DCB; len = DCB; }
  float acc = 0.0f;
  for (int i = threadIdx.x; i < len; i += 256) { float x = v[i]; acc += x * x; }
  float tot = block_sum(acc, s);
  float scale = g[row] / sqrtf(tot);
  for (int i = threadIdx.x; i < len; i += 256) w[i] = v[i] * scale;
}

// ------------------------------------------------ normalize codebook
__global__ void vq_prep_cbn(const float* cb, float* cbn, float* cbsq) {
  __shared__ float s[8];
  int row = blockIdx.x;
  float x = cb[(size_t)row * DCB + threadIdx.x];
  float tot = block_sum(x * x, s);
  float nv = fmaxf(sqrtf(tot), 1e-12f);
  cbn[(size_t)row * DCB + threadIdx.x] = x / nv;
  if (threadIdx.x == 0) cbsq[row] = tot / (nv * nv);
}

// --------------------------- z_e = w_in @ z + in_b  (f32 WMMA 16x16x4)
// 4 M-tiles per wave: the strided z loads (B operand) feed 4 WMMAs each.
// Output stored in enc layout: enc[(b*T + t)*256 + o]
__global__ void vq_gemm_in(const float* __restrict__ w_in,
                           const float* __restrict__ z,
                           const float* __restrict__ in_b,
                           float* __restrict__ enc) {
  int wave = threadIdx.x >> 5, lane = threadIdx.x & 31;
  int g = blockIdx.x * 8 + wave;                 // 4096 wave jobs
  int b = g >> 8;
  int rem = g & 255;
  int mg = rem >> 6, nt = rem & 63;
  int m0 = mg * 64, t0 = nt * 16;
  int half = lane >> 4, n = lane & 15, m = lane & 15;

  v8f acc0 = {}, acc1 = {}, acc2 = {}, acc3 = {};
  const float* A = w_in + (size_t)(m0 + m) * DIN;
  const float* Zb = z + ((size_t)b * DIN) * TT + t0 + n;
  for (int k = 0; k < DIN; k += 4) {
    int kk = k + half * 2;
    v2f bv;
    bv.x = Zb[(size_t)kk * TT];
    bv.y = Zb[(size_t)(kk + 1) * TT];
    v2f a0 = *(const v2f*)(A + 0 * 16 * DIN + kk);
    v2f a1 = *(const v2f*)(A + 1 * 16 * DIN + kk);
    v2f a2 = *(const v2f*)(A + 2 * 16 * DIN + kk);
    v2f a3 = *(const v2f*)(A + 3 * 16 * DIN + kk);
    acc0 = __builtin_amdgcn_wmma_f32_16x16x4_f32(false, a0, false, bv, (short)0, acc0, false, false);
    acc1 = __builtin_amdgcn_wmma_f32_16x16x4_f32(false, a1, false, bv, (short)0, acc1, false, false);
    acc2 = __builtin_amdgcn_wmma_f32_16x16x4_f32(false, a2, false, bv, (short)0, acc2, false, false);
    acc3 = __builtin_amdgcn_wmma_f32_16x16x4_f32(false, a3, false, bv, (short)0, acc3, false, false);
  }
  int t = t0 + n;
  float* erow = enc + ((size_t)b * TT + t) * DCB;
  for (int v = 0; v < 8; ++v) {
    int o0 = m0 + 0 * 16 + v + half * 8;
    int o1 = m0 + 1 * 16 + v + half * 8;
    int o2 = m0 + 2 * 16 + v + half * 8;
    int o3 = m0 + 3 * 16 + v + half * 8;
    erow[o0] = acc0[v] + in_b[o0];
    erow[o1] = acc1[v] + in_b[o1];
    erow[o2] = acc2[v] + in_b[o2];
    erow[o3] = acc3[v] + in_b[o3];
  }
}

// ------------------------------------------------ row-normalize enc
__global__ void vq_norm_enc(const float* enc, float* encn, float* encsq) {
  __shared__ float s[8];
  int row = blockIdx.x;
  float x = enc[(size_t)row * DCB + threadIdx.x];
  float tot = block_sum(x * x, s);
  float nv = fmaxf(sqrtf(tot), 1e-12f);
  encn[(size_t)row * DCB + threadIdx.x] = x / nv;
  if (threadIdx.x == 0) encsq[row] = tot / (nv * nv);
}

// ------- dist = encsq - 2*encn@cbn^T + cbsq ; write dist; argmin -> idx
// Enc tile (16 rows x 256) DMA'd into LDS by the Tensor Data Mover; each
// wave register-blocks 4 column tiles so one LDS A-fetch feeds 4 WMMAs.
__global__ void vq_dist_argmin(const float* __restrict__ encn,
                               const float* __restrict__ encsq,
                               const float* __restrict__ cbn,
                               const float* __restrict__ cbsq,
                               float* __restrict__ dist_out,
                               int* __restrict__ ws_idx,
                               float* __restrict__ idx_out,
                               int* __restrict__ cnt) {
  __shared__ float sA[16 * DCB];
  __shared__ float sEsq[16];
  __shared__ float sMinV[8][16];
  __shared__ int   sMinI[8][16];

  int wave = threadIdx.x >> 5, lane = threadIdx.x & 31;
  int r0 = blockIdx.x * 16;

  // TDM: 2D tile load, 16 rows of 256 dwords, row stride 256 dwords.
  if (wave == 0) {
    unsigned long long ga = (unsigned long long)(const void*)(encn + (size_t)r0 * DCB);
    unsigned ldsb = (unsigned)(unsigned long long)(void*)sA;   // LDS byte offset
    v4u g0;
    g0.x = 1u;                                        // count=1, user mode
    g0.y = ldsb;                                      // lds_addr
    g0.z = (unsigned)ga;                              // global_addr[31:0]
    g0.w = (unsigned)((ga >> 32) & 0x01FFFFFFull) | 0x80000000u; // addr[56:32] | type=2
    v8u g1;
    g1[0] = 0x20000u;          // data_size=2 (4B), no multicast/pad/iterate
    g1[1] = 256u << 16;        // tensor_dim0 = 256 (bits 79:48)
    g1[2] = 16384u << 16;      // tensor_dim1 = 16384 (bits 111:80)
    g1[3] = 256u << 16;        // tile_dim0 = 256 (bits 127:112)
    g1[4] = 16u;               // tile_dim1 = 16
    g1[5] = 256u;              // tensor_dim0_stride = 256
    g1[6] = 0u;
    g1[7] = 0u;
    asm volatile("tensor_load_to_lds %0, %1\n\t"
                 "s_wait_tensorcnt 0x0"
                 :: "s"(g0), "s"(g1) : "memory");
  }
  if (threadIdx.x < 16) sEsq[threadIdx.x] = encsq[r0 + threadIdx.x];
  __syncthreads();

  int half = lane >> 4, n = lane & 15, m = lane & 15;
  const float* Ap = sA + m * DCB;

  float minv[8];
  int mini[8];
  for (int v = 0; v < 8; ++v) { minv[v] = FLT_MAX; mini[v] = INT_MAX; }

  for (int ct = wave; ct < KK / 64; ct += 8) {    // 128 groups of 4 col tiles
    int c0 = ct * 64;
    const float* Bp0 = cbn + ((size_t)(c0 + 0 * 16 + n)) * DCB;
    const float* Bp1 = cbn + ((size_t)(c0 + 1 * 16 + n)) * DCB;
    const float* Bp2 = cbn + ((size_t)(c0 + 2 * 16 + n)) * DCB;
    const float* Bp3 = cbn + ((size_t)(c0 + 3 * 16 + n)) * DCB;
    v8f acc0 = {}, acc1 = {}, acc2 = {}, acc3 = {};
    for (int k = 0; k < DCB; k += 4) {
      int kk = k + half * 2;
      v2f a = *(const v2f*)(Ap + kk);     // one LDS b64 feeds 4 WMMAs
      v2f b0 = *(const v2f*)(Bp0 + kk);   // codebook: L2-resident (8MB/192MB)
      v2f b1 = *(const v2f*)(Bp1 + kk);
      v2f b2 = *(const v2f*)(Bp2 + kk);
      v2f b3 = *(const v2f*)(Bp3 + kk);
      acc0 = __builtin_amdgcn_wmma_f32_16x16x4_f32(false, a, false, b0, (short)0, acc0, false, false);
      acc1 = __builtin_amdgcn_wmma_f32_16x16x4_f32(false, a, false, b1, (short)0, acc1, false, false);
      acc2 = __builtin_amdgcn_wmma_f32_16x16x4_f32(false, a, false, b2, (short)0, acc2, false, false);
      acc3 = __builtin_amdgcn_wmma_f32_16x16x4_f32(false, a, false, b3, (short)0, acc3, false, false);
    }
    for (int j = 0; j < 4; ++j) {
      int col = c0 + j * 16 + n;
      float csq = cbsq[col];
      const v8f* accp = (j == 0) ? &acc0 : (j == 1) ? &acc1 : (j == 2) ? &acc2 : &acc3;
      for (int v = 0; v < 8; ++v) {
        int rl = v + half * 8;
        float d = sEsq[rl] - 2.0f * (*accp)[v] + csq;
        dist_out[((size_t)(r0 + rl)) * KK + col] = d;
        if (d < minv[v] || (d == minv[v] && col < mini[v])) { minv[v] = d; mini[v] = col; }
      }
    }
  }

  // reduce across the 16 lanes holding the same rows
  for (int v = 0; v < 8; ++v) {
    float mv = minv[v];
    int mi = mini[v];
    for (int off = 8; off; off >>= 1) {
      float ov = __shfl_xor(mv, off, 16);
      int oi = __shfl_xor(mi, off, 16);
      if (ov < mv || (ov == mv && oi < mi)) { mv = ov; mi = oi; }
    }
    if ((lane & 15) == 0) {
      int rl = v + half * 8;
      sMinV[wave][rl] = mv;
      sMinI[wave][rl] = mi;
    }
  }
  __syncthreads();

  if (threadIdx.x < 16) {
    float bv = sMinV[0][threadIdx.x];
    int bi = sMinI[0][threadIdx.x];
    for (int w = 1; w < 8; ++w) {
      float ov = sMinV[w][threadIdx.x];
      int oi = sMinI[w][threadIdx.x];
      if (ov < bv || (ov == bv && oi < bi)) { bv = ov; bi = oi; }
    }
    int grow = r0 + threadIdx.x;
    ws_idx[grow] = bi;
    idx_out[grow] = (float)bi;
    atomicAdd(cnt + bi, 1);
  }
}

// ------------------------- gather z_q, accumulate per-batch MSE
__global__ void vq_gather_loss(const float* __restrict__ codebook,
                               const float* __restrict__ enc,
                               const int* __restrict__ ws_idx,
                               float* __restrict__ zq,
                               float* __restrict__ mse) {
  __shared__ float s[8];
  int bt = blockIdx.x;
  int b = bt >> 10;
  int idx = ws_idx[bt];
  float q = codebook[(size_t)idx * DCB + threadIdx.x];
  float e = enc[(size_t)bt * DCB + threadIdx.x];
  zq[(size_t)bt * DCB + threadIdx.x] = q;
  float d = e - q;
  float tot = block_sum(d * d, s);
  if (threadIdx.x == 0) atomicAdd(mse + b, tot);
}

// --------------------- z_q_out = w_out @ z_q + out_b (f32 WMMA)
// 4 M-tiles per wave: one z_q B-fetch feeds 4 WMMAs.
__global__ void vq_gemm_out(const float* __restrict__ w_out,
                            const float* __restrict__ zq,
                            const float* __restrict__ out_b,
                            float* __restrict__ out) {
  int wave = threadIdx.x >> 5, lane = threadIdx.x & 31;
  int g = blockIdx.x * 8 + wave;                 // 16384 wave jobs
  int b = g >> 10;
  int rem = g & 1023;
  int mg = rem >> 6, nt = rem & 63;
  int m0 = mg * 64, t0 = nt * 16;
  int half = lane >> 4, n = lane & 15, m = lane & 15;

  v8f acc0 = {}, acc1 = {}, acc2 = {}, acc3 = {};
  const float* A = w_out + (size_t)(m0 + m) * DCB;
  const float* Bp = zq + ((size_t)b * TT + t0 + n) * DCB;
  for (int k = 0; k < DCB; k += 4) {
    int kk = k + half * 2;
    v2f bv = *(const v2f*)(Bp + kk);
    v2f a0 = *(const v2f*)(A + 0 * 16 * DCB + kk);
    v2f a1 = *(const v2f*)(A + 1 * 16 * DCB + kk);
    v2f a2 = *(const v2f*)(A + 2 * 16 * DCB + kk);
    v2f a3 = *(const v2f*)(A + 3 * 16 * DCB + kk);
    acc0 = __builtin_amdgcn_wmma_f32_16x16x4_f32(false, a0, false, bv, (short)0, acc0, false, false);
    acc1 = __builtin_amdgcn_wmma_f32_16x16x4_f32(false, a1, false, bv, (short)0, acc1, false, false);
    acc2 = __builtin_amdgcn_wmma_f32_16x16x4_f32(false, a2, false, bv, (short)0, acc2, false, false);
    acc3 = __builtin_amdgcn_wmma_f32_16x16x4_f32(false, a3, false, bv, (short)0, acc3, false, false);
  }
  int t = t0 + n;
  for (int v = 0; v < 8; ++v) {
    int o0 = m0 + 0 * 16 + v + half * 8;
    int o1 = m0 + 1 * 16 + v + half * 8;
    int o2 = m0 + 2 * 16 + v + half * 8;
    int o3 = m0 + 3 * 16 + v + half * 8;
    out[((size_t)b * DIN + o0) * TT + t] = acc0[v] + out_b[o0];
    out[((size_t)b * DIN + o1) * TT + t] = acc1[v] + out_b[o1];
    out[((size_t)b * DIN + o2) * TT + t] = acc2[v] + out_b[o2];
    out[((size_t)b * DIN + o3) * TT + t] = acc3[v] + out_b[o3];
  }
}

// ------------------------------- perplexity, active_num, losses
__global__ void vq_finalize(const int* __restrict__ cnt,
                            const float* __restrict__ cluster_size,
                            const float* __restrict__ mse,
                            float* __restrict__ d_out) {
  __shared__ float s[8];
  float ent = 0.0f, act = 0.0f;
  for (int k = threadIdx.x; k < KK; k += 256) {
    float c = (float)cnt[k];
    float avg = c * (1.0f / (float)NTOK);
    ent += avg * logf(avg + 1e-10f);
    float ncs = cluster_size[k] * 0.99f + c * 0.01f;
    act += (ncs > 2.0f) ? 1.0f : 0.0f;
  }
  float ent_t = block_sum(ent, s);
  float act_t = block_sum(act, s);
  if (threadIdx.x == 0) {
    d_out[OFF_PERP] = expf(-ent_t);
    d_out[OFF_ACT] = act_t;
  }
  if (threadIdx.x < BB) {
    float m = mse[threadIdx.x] * (1.0f / (float)(DCB * TT));
    d_out[OFF_COMMIT + threadIdx.x] = m * 0.15f;
    d_out[OFF_CBLOSS + threadIdx.x] = m;
  }
}

extern "C" void kernel_launch(void* const* d_in, const int* in_sizes, int n_in,
                              void* d_out, int out_size, void* d_ws, size_t ws_size,
                              hipStream_t stream) {
  const float* z            = (const float*)d_in[0];
  const float* in_g         = (const float*)d_in[1];
  const float* in_v         = (const float*)d_in[2];
  const float* in_b         = (const float*)d_in[3];
  const float* out_g        = (const float*)d_in[4];
  const float* out_v        = (const float*)d_in[5];
  const float* out_b        = (const float*)d_in[6];
  const float* codebook     = (const float*)d_in[7];
  const float* cluster_size = (const float*)d_in[8];

  float* ws = (float*)d_ws;
  float* w_in  = ws + WS_WIN;
  float* w_out = ws + WS_WOUT;
  float* cbn   = ws + WS_CBN;
  float* cbsq  = ws + WS_CBSQ;
  float* enc   = ws + WS_ENC;
  float* encn  = ws + WS_ENCN;
  float* encsq = ws + WS_ENCSQ;
  float* zq    = ws + WS_ZQ;
  float* mse   = ws + WS_MSE;
  int*   widx  = (int*)(ws + WS_IDX);
  int*   cnt   = (int*)(ws + WS_CNT);

  float* out = (float*)d_out;

  vq_zero<<<(KK + 255) / 256, 256, 0, stream>>>(cnt, mse);
  vq_prep_weights<<<DCB + DIN, 256, 0, stream>>>(in_g, in_v, out_g, out_v, w_in, w_out);
  vq_prep_cbn<<<KK, 256, 0, stream>>>(codebook, cbn, cbsq);
  vq_gemm_in<<<512, 256, 0, stream>>>(w_in, z, in_b, enc);
  vq_norm_enc<<<NTOK, 256, 0, stream>>>(enc, encn, encsq);
  vq_dist_argmin<<<NTOK / 16, 256, 0, stream>>>(encn, encsq, cbn, cbsq,
                                                out + OFF_DIST, widx,
                                                out + OFF_IDX, cnt);
  vq_gather_loss<<<NTOK, 256, 0, stream>>>(codebook, enc, widx, zq, mse);
  vq_gemm_out<<<2048, 256, 0, stream>>>(w_out, zq, out_b, out + OFF_ZQOUT);
  vq_finalize<<<1, 256, 0, stream>>>(cnt, cluster_size, mse, out);
}